// dilation_dev_63720134804151
// MI455X (gfx1250) — compile-verified
//
#include <hip/hip_runtime.h>

typedef float v2f __attribute__((ext_vector_type(2)));
typedef float v4f __attribute__((ext_vector_type(4)));
typedef float v8f __attribute__((ext_vector_type(8)));

#define BATCH 32
#define TLEN  2048
#define CH    64
#define HD    12
#define SEG   2048
#define OUTW  (HD*HD + CH)          // 208
#define NSEG  (BATCH * SEG)         // 65536
#define WPB   8                     // waves per block (wave32)
#define TPB   (WPB * 32)

// ---- LDS layout (in floats) ----
// AskP[c][w:16][h:20] = A[c,h,w] - A[c,w,h] for h<12 && w<12, else 0.
// w padded to 16 -> all lanes load branchlessly; h stride 20 -> only 2-way
// bank conflicts on the per-lane b128 pair (floor for 32-lane b128).
#define ASK_H  20
#define ASK_CS (16 * ASK_H)         // 320 floats per channel
#define ASK_F  (CH * ASK_CS)        // 20480 floats = 80 KB
// per-wave arena:
#define AR_W   0                    // window taps W[8][64]
#define AR_DX  512                  // DX[7][64] (pre-scaled by 2^-6)
#define AR_M   (512 + 448)          // 16x16 Ms tile
#define AR_P   (AR_M + 256)         // 16x16 P tile
#define AR_Z   (AR_P + 256)         // 16x16 Z tile
#define AR_F   (AR_Z + 256)         // 1728 floats / wave
#define SMEM_F (ASK_F + WPB * AR_F) // 34304 floats = 137216 bytes

extern "C" __global__ __launch_bounds__(TPB)
void seg_rot_expm_kernel(const float* __restrict__ x,
                         const float* __restrict__ A,
                         float* __restrict__ out)
{
    extern __shared__ float sm[];
    const int tid  = threadIdx.x;
    const int lane = tid & 31;
    const int wid  = tid >> 5;
    const int col  = lane & 15;     // N index of this lane's C-fragment column
    const int hi   = lane >> 4;     // half-wave: rows v+8*hi

    // ---------- build padded skew tensor AskP (whole block, once) ----------
    for (int i = tid; i < ASK_F; i += TPB) {
        int h  = i % ASK_H;
        int cw = i / ASK_H;
        int w  = cw & 15;
        int c  = cw >> 4;
        float v = 0.0f;
        if (h < HD && w < HD)
            v = A[c*HD*HD + h*HD + w] - A[c*HD*HD + w*HD + h];
        sm[i] = v;
    }

    float* ar = sm + ASK_F + wid * AR_F;
    const int n = blockIdx.x * WPB + wid;   // segment id; grid covers exactly NSEG
    const int b = n >> 11;
    const int s = n & (SEG - 1);

    __syncthreads();

    // ---------- load dilated window taps W[8][64] (edge clamped) ----------
    {
        const int k = lane & 7;             // tap
        const int q = lane >> 3;            // channel quarter (16 ch)
        int t = s - 7 + k * 2;
        t = t < 0 ? 0 : (t > TLEN-1 ? TLEN-1 : t);
        const v4f* src = (const v4f*)(x + ((size_t)(b * TLEN + t)) * CH);
        v4f* dst = (v4f*)(ar + AR_W + k * CH + q * 16);
        #pragma unroll
        for (int j = 0; j < 4; ++j) dst[j] = src[q*4 + j];
    }
    __syncthreads();

    // ---------- DX[k][c] = (W[k+1][c]-W[k][c]) * 2^-6 (pre-scaled for expm) ----------
    {
        const int k = lane & 7;
        const int q = lane >> 3;
        if (k < 7) {
            #pragma unroll
            for (int j = 0; j < 4; ++j) {
                v4f a0 = *(const v4f*)(ar + AR_W + (k+1)*CH + q*16 + j*4);
                v4f b0 = *(const v4f*)(ar + AR_W +  k   *CH + q*16 + j*4);
                *(v4f*)(ar + AR_DX + k*CH + q*16 + j*4) = (a0 - b0) * 0.015625f;
            }
        }
    }
    __syncthreads();

    // branchless diagonal indicator: dsel[v]=1 iff (v+8*hi)==col
    float dsel[8];
    #pragma unroll
    for (int v = 0; v < 8; ++v) dsel[v] = ((v + 8*hi) == col) ? 1.0f : 0.0f;

    const float cfs[9] = {1.f, 1.f, 0.5f, 1.f/6.f, 1.f/24.f,
                          1.f/120.f, 1.f/720.f, 1.f/5040.f, 1.f/40320.f};

    float zreg[8];
    #pragma unroll
    for (int v = 0; v < 8; ++v) zreg[v] = 0.f;

    for (int k = 0; k < 7; ++k) {
        // ---- Ms = sum_c dx[k][c]*Ask[c], built directly in C-fragment layout ----
        float mreg[8];
        #pragma unroll
        for (int v = 0; v < 8; ++v) mreg[v] = 0.f;
        {
            const float* askb = sm + col*ASK_H + hi*8;    // + c*ASK_CS
            const float* dxk  = ar + AR_DX + k*CH;
            for (int c = 0; c < CH; c += 4) {
                v4f d4 = *(const v4f*)(dxk + c);          // wave-uniform broadcast
                #pragma unroll
                for (int j = 0; j < 4; ++j) {
                    v4f a0 = *(const v4f*)(askb + (c+j)*ASK_CS);
                    v4f a1 = *(const v4f*)(askb + (c+j)*ASK_CS + 4);
                    float d = d4[j];
                    mreg[0] += d*a0.x; mreg[1] += d*a0.y;
                    mreg[2] += d*a0.z; mreg[3] += d*a0.w;
                    mreg[4] += d*a1.x; mreg[5] += d*a1.y;
                    mreg[6] += d*a1.z; mreg[7] += d*a1.w;
                }
            }
        }

        // ---- stage Ms tile (row-major 16x16) and keep its B-fragments ----
        #pragma unroll
        for (int v = 0; v < 8; ++v)
            ar[AR_M + (v + 8*hi)*16 + col] = mreg[v];
        v2f bm[3];
        #pragma unroll
        for (int q = 0; q < 3; ++q) {
            bm[q].x = ar[AR_M + (4*q + 2*hi    )*16 + col];
            bm[q].y = ar[AR_M + (4*q + 2*hi + 1)*16 + col];
        }

        // ---- Horner seed: P = c8*Ms + c7*I ----
        float preg[8];
        #pragma unroll
        for (int v = 0; v < 8; ++v) preg[v] = cfs[8]*mreg[v] + cfs[7]*dsel[v];

        // ---- Horner: P = P*Ms + c_j*I, j = 6..0  (K=12 -> 3 wmma chunks) ----
        #pragma unroll
        for (int j = 6; j >= 0; --j) {
            #pragma unroll
            for (int v = 0; v < 8; ++v)
                ar[AR_P + (v + 8*hi)*16 + col] = preg[v];
            v8f c = {};
            #pragma unroll
            for (int q = 0; q < 3; ++q) {
                v2f a = *(const v2f*)(ar + AR_P + col*16 + 4*q + 2*hi);
                c = __builtin_amdgcn_wmma_f32_16x16x4_f32(false, a, false, bm[q],
                                                          (short)0, c, false, false);
            }
            #pragma unroll
            for (int v = 0; v < 8; ++v) preg[v] = c[v] + cfs[j]*dsel[v];
        }

        // ---- 6 squarings: P = P*P  (exp(M) = (exp(M/64))^64) ----
        #pragma unroll
        for (int sq = 0; sq < 6; ++sq) {
            #pragma unroll
            for (int v = 0; v < 8; ++v)
                ar[AR_P + (v + 8*hi)*16 + col] = preg[v];
            v2f a[3], bb[3];
            #pragma unroll
            for (int q = 0; q < 3; ++q) {
                a[q]  = *(const v2f*)(ar + AR_P + col*16 + 4*q + 2*hi);
                bb[q].x = ar[AR_P + (4*q + 2*hi    )*16 + col];
                bb[q].y = ar[AR_P + (4*q + 2*hi + 1)*16 + col];
            }
            v8f c = {};
            #pragma unroll
            for (int q = 0; q < 3; ++q)
                c = __builtin_amdgcn_wmma_f32_16x16x4_f32(false, a[q], false, bb[q],
                                                          (short)0, c, false, false);
            #pragma unroll
            for (int v = 0; v < 8; ++v) preg[v] = c[v];
        }

        // ---- chain: z = z * E_k ----
        if (k == 0) {
            #pragma unroll
            for (int v = 0; v < 8; ++v) zreg[v] = preg[v];
        } else {
            #pragma unroll
            for (int v = 0; v < 8; ++v) {
                ar[AR_Z + (v + 8*hi)*16 + col] = zreg[v];
                ar[AR_P + (v + 8*hi)*16 + col] = preg[v];
            }
            v8f c = {};
            #pragma unroll
            for (int q = 0; q < 3; ++q) {
                v2f a = *(const v2f*)(ar + AR_Z + col*16 + 4*q + 2*hi);
                v2f bb;
                bb.x = ar[AR_P + (4*q + 2*hi    )*16 + col];
                bb.y = ar[AR_P + (4*q + 2*hi + 1)*16 + col];
                c = __builtin_amdgcn_wmma_f32_16x16x4_f32(false, a, false, bb,
                                                          (short)0, c, false, false);
            }
            #pragma unroll
            for (int v = 0; v < 8; ++v) zreg[v] = c[v];
        }
    }

    // ---------- outputs: [dev(144) | sp(64)] ----------
    float* orow = out + (size_t)n * OUTW;
    if (col < HD) {
        #pragma unroll
        for (int v = 0; v < 8; ++v) {
            int r = v + 8*hi;
            if (r < HD) orow[r*HD + col] = zreg[v];
        }
    }
    orow[HD*HD + lane]      = ar[AR_W + lane];        // tap 0, channels 0..31
    orow[HD*HD + 32 + lane] = ar[AR_W + 32 + lane];   // tap 0, channels 32..63
}

extern "C" void kernel_launch(void* const* d_in, const int* in_sizes, int n_in,
                              void* d_out, int out_size, void* d_ws, size_t ws_size,
                              hipStream_t stream) {
    (void)in_sizes; (void)n_in; (void)out_size; (void)d_ws; (void)ws_size;
    const float* x = (const float*)d_in[0];
    const float* A = (const float*)d_in[1];
    float* out     = (float*)d_out;

    const size_t smem = SMEM_F * sizeof(float);   // ~134 KB -> 2 blocks/WGP (320 KB)
    (void)hipFuncSetAttribute((const void*)seg_rot_expm_kernel,
                              hipFuncAttributeMaxDynamicSharedMemorySize,
                              (int)smem);
    seg_rot_expm_kernel<<<NSEG / WPB, TPB, smem, stream>>>(x, A, out);
}